// SemCHGraphConv_6751688589756
// MI455X (gfx1250) — compile-verified
//
#include <hip/hip_runtime.h>
#include <hip/hip_bf16.h>

typedef __attribute__((ext_vector_type(2))) float v2f;
typedef __attribute__((ext_vector_type(8))) float v8f;

#define NEGV -9.0e15f
#define FT 16   // f-tile per workgroup
#define BT 16   // batch tile per workgroup (WMMA N dim in phase 2)

__device__ __forceinline__ v8f wmma4(v2f a, v2f b, v8f c) {
  // D = A(16x4 f32) * B(4x16 f32) + C(16x16 f32)
  return __builtin_amdgcn_wmma_f32_16x16x4_f32(
      /*neg_a=*/false, a, /*neg_b=*/false, b,
      /*c_mod=*/(short)0, c, /*reuse_a=*/false, /*reuse_b=*/false);
}

// async global->LDS copy of 16B, tracked by ASYNCcnt
__device__ __forceinline__ void async_copy_b128(const float* gptr, float* lptr) {
  unsigned lds_addr =
      (unsigned)(size_t)(__attribute__((address_space(3))) float*)lptr;
  unsigned long long gaddr = (unsigned long long)(size_t)gptr;
  asm volatile("global_load_async_to_lds_b128 %0, %1, off"
               :: "v"(lds_addr), "v"(gaddr) : "memory");
}
__device__ __forceinline__ void async_wait0() {
  asm volatile("s_wait_asynccnt 0x0" ::: "memory");
}

// ---------------- K1: masked per-channel row softmax ----------------
// e:[F,64,64], mask:[64,64] (bool/u8) -> off:[F,64,64] (diag zeroed), d:[F,64]
__global__ __launch_bounds__(256) void softmax_prep(
    const float* __restrict__ e, const unsigned char* __restrict__ mask,
    float* __restrict__ off, float* __restrict__ dvec) {
  const int wave = threadIdx.x >> 5;
  const int lane = threadIdx.x & 31;
  const int row  = blockIdx.x * 8 + wave;     // row in [0, F*64)
  const int n    = row & 63;
  const float* er = e + (size_t)row * 64;
  const int j0 = lane, j1 = lane + 32;

  float v0 = mask[n * 64 + j0] ? er[j0] : NEGV;
  float v1 = mask[n * 64 + j1] ? er[j1] : NEGV;

  float m = fmaxf(v0, v1);
  #pragma unroll
  for (int s = 16; s > 0; s >>= 1) m = fmaxf(m, __shfl_xor(m, s, 32));
  float e0 = __expf(v0 - m), e1 = __expf(v1 - m);
  float sum = e0 + e1;
  #pragma unroll
  for (int s = 16; s > 0; s >>= 1) sum += __shfl_xor(sum, s, 32);
  float inv = 1.0f / sum;
  float a0 = e0 * inv, a1 = e1 * inv;

  if (j0 == n) { dvec[row] = a0; a0 = 0.0f; }   // d[f*64+n] = diagonal
  if (j1 == n) { dvec[row] = a1; a1 = 0.0f; }

  float* orow = off + (size_t)row * 64;
  orow[j0] = a0;
  orow[j1] = a1;
}

// ---------------- K2: fused GEMM + diag scale + per-channel einsum ----------------
// Grid: x = F/FT (fast, shares x in L2), y = B/BT. Block: 256 (8 waves).
// Dynamic LDS: wpan[2][256][FT] (32KB) | hbuf1[BT*64][FT] (64KB) | obuf[BT*64][FT] (64KB)
__global__ __launch_bounds__(256) void fused_gconv(
    const float* __restrict__ x, const float* __restrict__ W,
    const float* __restrict__ bias, const float* __restrict__ off,
    const float* __restrict__ dvec, float* __restrict__ out) {
  extern __shared__ float smem[];
  float* wpan  = smem;                 // 8192 floats
  float* hbuf1 = smem + 8192;          // 16384 floats
  float* obuf  = smem + 8192 + 16384;  // 16384 floats

  const int tid  = threadIdx.x;
  const int wave = tid >> 5;
  const int lane = tid & 31;
  const int hl   = lane >> 4;   // half-wave select (K sub-offset / M high bit)
  const int ll   = lane & 15;   // M row (A) or N col (B/D)

  const int f0 = blockIdx.x * FT;
  const size_t rowbase = (size_t)blockIdx.y * (BT * 64);  // flat (b,n) row base

  // Stage W0/W1 panels via async DMA (16B chunks; panel rows are 64B):
  // wpan[wsel*4096 + k*16 + fl] = W[wsel][k][f0+fl]
  for (int i = tid; i < 2048; i += 256) {     // 2048 x 16B chunks
    int fo = (i & 3) * 4;                     // f sub-offset 0,4,8,12
    int k  = (i >> 2) & 255;
    int ws = i >> 10;
    async_copy_b128(W + (size_t)ws * 65536 + (size_t)k * 256 + (f0 + fo),
                    wpan + i * 4);
  }
  async_wait0();
  __syncthreads();

  const float breg = bias[f0 + ll];

  // ---- Phase 1: H = X @ {W0,W1}, 1024 local rows x FT cols.
  // Two row-tiles per iteration: 4 independent WMMA chains, shared B loads.
  for (int rt = wave; rt < 32; rt += 8) {     // tiles rt and rt+32
    v8f acc0a = {}, acc1a = {};               // row-tile rt      (W0 / W1)
    v8f acc0b = {}, acc1b = {};               // row-tile rt + 32 (W0 / W1)
    const float* xpa = x + (rowbase + rt * 16 + ll) * 256 + 2 * hl;
    const float* xpb = xpa + 32 * 16 * 256;
    __builtin_prefetch(xpa + 8 * 16 * 256, 0, 0);  // warm next tiles
    __builtin_prefetch(xpb + 8 * 16 * 256, 0, 0);
    #pragma unroll 4
    for (int kk = 0; kk < 64; ++kk) {         // K = 256 in steps of 4
      v2f aa = *(const v2f*)(xpa + 4 * kk);
      v2f ab = *(const v2f*)(xpb + 4 * kk);
      int bi = (4 * kk + 2 * hl) * FT + ll;   // B: K rows striped, N=f
      v2f b0, b1;
      b0.x = wpan[bi];        b0.y = wpan[bi + FT];
      b1.x = wpan[4096 + bi]; b1.y = wpan[4096 + bi + FT];
      acc0a = wmma4(aa, b0, acc0a);
      acc0b = wmma4(ab, b0, acc0b);
      acc1a = wmma4(aa, b1, acc1a);
      acc1b = wmma4(ab, b1, acc1b);
    }
    // Epilogue: obuf = d[f,n]*h0 + bias ; hbuf1 = h1
    #pragma unroll
    for (int v = 0; v < 8; ++v) {
      int rla = rt * 16 + v + 8 * hl;         // local flat (b,n) row
      int rlb = rla + 32 * 16;
      float dva = dvec[(f0 + ll) * 64 + (rla & 63)];
      float dvb = dvec[(f0 + ll) * 64 + (rlb & 63)];
      obuf [rla * FT + ll] = acc0a[v] * dva + breg;
      obuf [rlb * FT + ll] = acc0b[v] * dvb + breg;
      hbuf1[rla * FT + ll] = acc1a[v];
      hbuf1[rlb * FT + ll] = acc1b[v];
    }
  }
  __syncthreads();

  // ---- Phase 2: per channel f, Out(64n x 16b) += off_f(64x64) @ h1_f(64m x 16b).
  // Two n-tiles per iteration at same f: shared B loads, 2 WMMA chains.
  for (int job = wave; job < 32; job += 8) {  // FT channels x 2 nt-pairs
    const int fl  = job & 15;
    const int ntp = job >> 4;                 // 0 or 1 -> n-tiles 2*ntp, 2*ntp+1
    v8f acca = {}, accb = {};
    const float* offpa =
        off + ((size_t)(f0 + fl) * 64 + ntp * 32 + ll) * 64 + 2 * hl;
    const float* offpb = offpa + 16 * 64;
    #pragma unroll 4
    for (int kk = 0; kk < 16; ++kk) {         // K = 64 (m) in steps of 4
      v2f aa = *(const v2f*)(offpa + 4 * kk); // A: M=n rows of off_f
      v2f ab = *(const v2f*)(offpb + 4 * kk);
      int m  = 4 * kk + 2 * hl;
      int bi = (ll * 64 + m) * FT + fl;       // B: K=m rows, N=b cols
      v2f b;
      b.x = hbuf1[bi];
      b.y = hbuf1[bi + FT];
      acca = wmma4(aa, b, acca);
      accb = wmma4(ab, b, accb);
    }
    #pragma unroll
    for (int v = 0; v < 8; ++v) {
      int na = ntp * 32 + v + 8 * hl;
      int nb = na + 16;
      atomicAdd(&obuf[(ll * 64 + na) * FT + fl], acca[v]);  // ds_add_f32
      atomicAdd(&obuf[(ll * 64 + nb) * FT + fl], accb[v]);
    }
  }
  __syncthreads();

  // ---- Coalesced output: 64B per (b,n) row at f0 (64B aligned) ----
  for (int r = tid; r < BT * 64; r += 256) {
    const float4* src = (const float4*)(obuf + r * FT);
    float4 t0 = src[0], t1 = src[1], t2 = src[2], t3 = src[3];
    float* op = out + (rowbase + r) * 256 + f0;
    ((float4*)op)[0] = t0;
    ((float4*)op)[1] = t1;
    ((float4*)op)[2] = t2;
    ((float4*)op)[3] = t3;
  }
}

extern "C" void kernel_launch(void* const* d_in, const int* in_sizes, int n_in,
                              void* d_out, int out_size, void* d_ws, size_t ws_size,
                              hipStream_t stream) {
  const float*         x    = (const float*)d_in[0];          // [2048,64,256]
  const float*         W    = (const float*)d_in[1];          // [2,256,256]
  const float*         e    = (const float*)d_in[2];          // [256,64,64]
  const float*         bias = (const float*)d_in[3];          // [256]
  const unsigned char* mask = (const unsigned char*)d_in[4];  // [64,64] bool
  float* out = (float*)d_out;

  // Workspace: off [256*64*64] floats (4MB) then d [256*64] floats (64KB)
  float* off  = (float*)d_ws;
  float* dvec = off + 256 * 64 * 64;

  // K1: 16384 softmax rows, 8 rows (waves) per block
  softmax_prep<<<dim3(2048), dim3(256), 0, stream>>>(e, mask, off, dvec);

  // K2: fused; f-tile fast so same-b blocks share x in L2
  const size_t lds_bytes = (8192 + 16384 + 16384) * sizeof(float);  // 160KB
  fused_gconv<<<dim3(256 / FT, 2048 / BT), dim3(256), lds_bytes, stream>>>(
      x, W, bias, off, dvec, out);
}